// CustomLossFilter_1_28827820491183
// MI455X (gfx1250) — compile-verified
//
#include <hip/hip_runtime.h>

// ---------------------------------------------------------------------------
// CustomLossFilter on MI455X (gfx1250, wave32)
//
// Memory-bound streaming reduction:
//   480 MB read @ 23.3 TB/s  =>  ~21 us floor. Compute is negligible.
// Grid-stride loop with coalesced b32 loads + speculative global_prefetch_b8,
// deterministic two-pass reduction. Cross-lane reduce uses
// V_WMMA_F32_16X16X4_F32 (ones-matrix trick) + one ds_swizzle (SWAPX16).
// ---------------------------------------------------------------------------

typedef float v2f __attribute__((ext_vector_type(2)));
typedef float v8f __attribute__((ext_vector_type(8)));

#define THREADS 256
#define NWAVES  (THREADS / 32)

// Sum s across all 32 lanes of a wave using one WMMA.
//   A (16x4 f32): lane L puts s in its K-slot {0 or 2}, 0 in {1 or 3}
//   B (4x16 f32) = all ones
//   D[m][n] = s_m + s_{m+16}  (independent of n)
// Lanes 0-15 hold D rows 0..7, lanes 16-31 rows 8..15; summing the 8
// accumulator VGPRs gives half-totals; one SWAPX16 swizzle completes the sum.
// Requires EXEC == all ones (call from fully-converged code only).
__device__ __forceinline__ float wave_reduce_wmma(float s) {
    v2f a; a.x = s;    a.y = 0.0f;
    v2f b; b.x = 1.0f; b.y = 1.0f;
    v8f c = {};
    // 8 args: (neg_a, A, neg_b, B, c_mod, C, reuse_a, reuse_b)
    c = __builtin_amdgcn_wmma_f32_16x16x4_f32(
        false, a, false, b, (short)0, c, false, false);
    float t = c[0] + c[1] + c[2] + c[3] + c[4] + c[5] + c[6] + c[7];
    // ds_swizzle_b32, group-of-32: xor_mask=0x10, or=0, and=0x1f  (SWAPX16)
    int sw = __builtin_amdgcn_ds_swizzle(__float_as_int(t), 0x401F);
    return t + __int_as_float(sw);   // full 32-lane sum, in every lane
}

// Block-level reduction for a 256-thread (8-wave) block.
// Result valid in threadIdx.x == 0.
__device__ __forceinline__ float block_reduce(float s) {
    __shared__ float wsum[NWAVES];
    const int lane = threadIdx.x & 31;
    const int wid  = threadIdx.x >> 5;

    const float w = wave_reduce_wmma(s);
    if (lane == 0) wsum[wid] = w;
    __syncthreads();

    float total = 0.0f;
    if (threadIdx.x == 0) {
#pragma unroll
        for (int i = 0; i < NWAVES; ++i) total += wsum[i];   // fixed order
    }
    return total;
}

__global__ void __launch_bounds__(THREADS)
loss_partial_kernel(const float* __restrict__ in,   // (N,5) row-major
                    const float* __restrict__ tg,   // (N,1)
                    float* __restrict__ partial,    // (gridDim.x)
                    int n) {
    const int tid    = blockIdx.x * blockDim.x + threadIdx.x;
    const int stride = gridDim.x * blockDim.x;

    float s = 0.0f;
    for (int i = tid; i < n; i += stride) {
        const size_t base = (size_t)i * 5u;
        // Unconditional speculative prefetch of the next grid-stride tile
        // (global_prefetch_b8; invalid translations are silently dropped).
        __builtin_prefetch(in + base + (size_t)stride * 5u, 0, 3);

        const float a0 = in[base + 0];
        const float a2 = in[base + 2];
        const float a4 = in[base + 4];
        const float t  = tg[i];

        const float d = (fabsf(a4 - a2) < 0.1f) ? (a0 - a4) : (a0 - t);
        s += fabsf(d);
    }

    // All lanes converged here; full EXEC for the WMMA reduce.
    const float total = block_reduce(s);
    if (threadIdx.x == 0) partial[blockIdx.x] = total;
}

__global__ void __launch_bounds__(THREADS)
final_reduce_kernel(const float* __restrict__ partial, int p,
                    float* __restrict__ out, float inv_n) {
    float s = 0.0f;
    for (int i = threadIdx.x; i < p; i += blockDim.x) s += partial[i];

    const float total = block_reduce(s);
    if (threadIdx.x == 0) out[0] = total * inv_n;   // mean; also inits d_out
}

extern "C" void kernel_launch(void* const* d_in, const int* in_sizes, int n_in,
                              void* d_out, int out_size, void* d_ws, size_t ws_size,
                              hipStream_t stream) {
    const float* in = (const float*)d_in[0];   // inputs  (N*5 f32)
    const float* tg = (const float*)d_in[1];   // targets (N   f32)
    float* out      = (float*)d_out;           // 1 f32
    float* partial  = (float*)d_ws;

    const int n = in_sizes[1];                 // N rows (targets is N*1)

    int blocks = 2048;                         // ~38 rows/thread at N=20M
    if ((size_t)blocks * sizeof(float) > ws_size)
        blocks = (int)(ws_size / sizeof(float));
    if (blocks < 1) blocks = 1;

    loss_partial_kernel<<<blocks, THREADS, 0, stream>>>(in, tg, partial, n);
    final_reduce_kernel<<<1, THREADS, 0, stream>>>(partial, blocks, out,
                                                   1.0f / (float)n);
}